// QLSTMGen025_65481071410781
// MI455X (gfx1250) — compile-verified
//
#include <hip/hip_runtime.h>
#include <hip/hip_bf16.h>

// QLSTM on MI455X (gfx1250).
//
// Strategy (latency-bound sequential scan, batch-parallel recurrence):
//   - 16 blocks x 256 threads; block i owns batches [16i, 16i+16) for all 512
//     steps (LSTM state is independent per batch element -> no inter-block sync).
//   - Quantum layer replaced by its closed form:
//       RX(x)RX(th) = RX(x+th); Heisenberg-evolved Z through the CNOT ring:
//       ez = 0.99 * [c1c2c3, c0c1, c0c1c2, c0c1c2c3],  c_w = cos(x_w+th_w+b_w)
//   - Both GEMMs run on V_WMMA_F32_16X16X4_F32 (full f32 precision):
//       GEMM1: pre(16x16) = w_lin(16x384) x comb^T; every wave does 4 K-steps on
//              the x part (global_load_b64) and 8 K-steps on the hx part
//              (ds_load_b64) -- address spaces never mixed, so no FLAT loads and
//              no per-WMMA loadcnt+dscnt serialization.
//       GEMM2: gates(16x16 per h-tile/gate) = w_map x acts, bias as C-fragment
//   - hx kept in LDS (padded row stride 260 -> conflict-free WMMA B-frag loads),
//     cx kept in C-fragment-shaped VGPRs.

typedef float v2f __attribute__((ext_vector_type(2)));
typedef float v4f __attribute__((ext_vector_type(4)));
typedef float v8f __attribute__((ext_vector_type(8)));

static constexpr int S  = 512;
static constexpr int B  = 256;
static constexpr int D  = 128;
static constexpr int H  = 256;
static constexpr int NB = 16;            // batches per block
static constexpr int R  = 16;            // NG*NQ rows of GEMM1
static constexpr int DH = D + H;         // 384
static constexpr int NWAVES = 8;
static constexpr int KX = (D / 4) / NWAVES;  // 4 x-part K-steps per wave
static constexpr int KH = (H / 4) / NWAVES;  // 8 hx-part K-steps per wave
static constexpr int HXS = 260;          // padded hx row stride (floats)

__device__ __forceinline__ float sigmoidf_(float x) {
    return 1.0f / (1.0f + expf(-x));
}

__global__ __launch_bounds__(256, 1)
void qlstm_scan_kernel(const float* __restrict__ inputs,  // (S,B,D)
                       const float* __restrict__ w_lin,   // (4,4,384) == (16,384)
                       const float* __restrict__ b_lin,   // (16)
                       const float* __restrict__ w_map,   // (4,256,4)
                       const float* __restrict__ b_map,   // (4,256)
                       const float* __restrict__ theta,   // (16)
                       float* __restrict__ out)           // outputs | hx | cx
{
    __shared__ float hx_s[NB * HXS];              // 16 batches x 256 (padded)
    __shared__ float partial_s[NWAVES * 16 * 17]; // GEMM1 K-split partials (padded)
    __shared__ float pre_s[R * 16];               // pre[r][b_local]
    __shared__ float acts_s[4 * NB * 4];          // acts[g][b_local][q]
    __shared__ float thetaEff_s[R];               // theta + b_lin

    const int tid  = threadIdx.x;
    const int lane = tid & 31;
    const int wv   = tid >> 5;      // wave 0..7
    const int n16  = lane & 15;     // lane % 16
    const int hi   = lane >> 4;     // lane / 16
    const int gb0  = blockIdx.x * NB;

    // ---------------- setup ----------------
    if (tid < R) thetaEff_s[tid] = theta[tid] + b_lin[tid];
    for (int i = tid; i < NB * HXS; i += blockDim.x) hx_s[i] = 0.0f;

    // Per-lane fragment offsets (floats), per documented 16x16x4 f32 layouts:
    //  A(16x4): lane l, vgpr v -> A[l%16][2*(l/16)+v]
    //  B(4x16): lane l, vgpr v -> B[2*(l/16)+v][l%16]
    //  C(16x16): vgpr j, lane l -> C[j + 8*(l/16)][l%16]
    const int voff_a  = n16 * DH  + 2 * hi;   // into w_lin rows
    const int voff_x  = n16 * D   + 2 * hi;   // into x rows (local batch)
    const int voff_h  = n16 * HXS + 2 * hi;   // into hx_s rows
    const int voff_m  = n16 * 4   + 2 * hi;   // into w_map[g] rows (K=4)
    const int voff_ac = n16 * 4   + 2 * hi;   // into acts_s[g] rows (K=4)

    // cx C-fragments: wave wv owns h-tiles h0 = 32*wv + 16*jt, jt in {0,1}
    v8f cx0 = {};
    v8f cx1 = {};

    __syncthreads();

    for (int t = 0; t < S; ++t) {
        // -------- Phase A: pre = w_lin x comb^T, K split across 8 waves --------
        // Wave wv: x-part K-steps  kx = wv*KX + i  (i<4)  -> dims 4*kx
        //          hx-part K-steps kh = wv*KH + i  (i<8)  -> dims 128 + 4*kh
        {
            const float* xbase = inputs + ((size_t)t * B + gb0) * D;
            // prefetch next step's x slice (contiguous 8KB per block)
            if (t + 1 < S)
                __builtin_prefetch(inputs + ((size_t)(t + 1) * B + gb0) * D + tid * 8, 0, 3);

            v8f acc = {};

            // x part: pure global loads, issued before the WMMA chain
            {
                v2f a[KX], bb[KX];
                #pragma unroll
                for (int i = 0; i < KX; ++i) {
                    const int kx = wv * KX + i;
                    a[i]  = *(const v2f*)(w_lin + voff_a + 4 * kx);
                    bb[i] = *(const v2f*)(xbase + voff_x + 4 * kx);
                }
                #pragma unroll
                for (int i = 0; i < KX; ++i)
                    acc = __builtin_amdgcn_wmma_f32_16x16x4_f32(
                              false, a[i], false, bb[i], (short)0, acc, false, false);
            }

            // hx part: pure LDS loads, issued before the WMMA chain
            {
                v2f a[KH], bb[KH];
                #pragma unroll
                for (int i = 0; i < KH; ++i) {
                    const int kh = wv * KH + i;
                    a[i]  = *(const v2f*)(w_lin + voff_a + D + 4 * kh);
                    bb[i] = *(const v2f*)(&hx_s[voff_h + 4 * kh]);
                }
                #pragma unroll
                for (int i = 0; i < KH; ++i)
                    acc = __builtin_amdgcn_wmma_f32_16x16x4_f32(
                              false, a[i], false, bb[i], (short)0, acc, false, false);
            }

            #pragma unroll
            for (int j = 0; j < 8; ++j)
                partial_s[wv * 272 + (j + 8 * hi) * 17 + n16] = acc[j];
        }
        __syncthreads();

        // -------- reduce K-partials -> pre_s (one (r,b) per thread) --------
        {
            const int r = tid >> 4, b = tid & 15;
            float s = 0.0f;
            #pragma unroll
            for (int w = 0; w < NWAVES; ++w)
                s += partial_s[w * 272 + r * 17 + b];
            pre_s[r * 16 + b] = s;
        }
        __syncthreads();

        // -------- Phase B: closed-form quantum layer + activations --------
        if (tid < 64) {
            const int g = tid >> 4, b = tid & 15;
            float c[4];
            #pragma unroll
            for (int q = 0; q < 4; ++q) {
                const int r = g * 4 + q;
                c[q] = cosf(pre_s[r * 16 + b] + thetaEff_s[r]);
            }
            const float nf = 0.99f;                // (1 - depolarizing rate)
            const float e1 = c[0] * c[1];
            const float e2 = e1 * c[2];
            const float e3 = e2 * c[3];
            const float e0 = c[1] * c[2] * c[3];
            float a0, a1, a2, a3;
            if (g == 2) {                          // update gate -> tanh
                a0 = tanhf(nf * e0); a1 = tanhf(nf * e1);
                a2 = tanhf(nf * e2); a3 = tanhf(nf * e3);
            } else {                               // f, i, o -> sigmoid
                a0 = sigmoidf_(nf * e0); a1 = sigmoidf_(nf * e1);
                a2 = sigmoidf_(nf * e2); a3 = sigmoidf_(nf * e3);
            }
            v4f av = { a0, a1, a2, a3 };
            *(v4f*)&acts_s[(g * 16 + b) * 4] = av;
        }
        __syncthreads();

        // -------- Phase C: gates via WMMA (bias as C-frag) + cell update --------
        {
            v2f bf[4];
            #pragma unroll
            for (int g = 0; g < 4; ++g)
                bf[g] = *(const v2f*)(&acts_s[g * 64 + voff_ac]);

            #pragma unroll
            for (int jt = 0; jt < 2; ++jt) {
                const int h0 = 32 * wv + 16 * jt;
                v8f gate[4];
                #pragma unroll
                for (int g = 0; g < 4; ++g) {
                    v2f a = *(const v2f*)(w_map + g * (H * 4) + h0 * 4 + voff_m);
                    const float* bp = b_map + g * H + h0 + 8 * hi;
                    v4f blo = *(const v4f*)(bp);
                    v4f bhi = *(const v4f*)(bp + 4);
                    v8f cinit;
                    cinit[0] = blo[0]; cinit[1] = blo[1];
                    cinit[2] = blo[2]; cinit[3] = blo[3];
                    cinit[4] = bhi[0]; cinit[5] = bhi[1];
                    cinit[6] = bhi[2]; cinit[7] = bhi[3];
                    gate[g] = __builtin_amdgcn_wmma_f32_16x16x4_f32(
                                  false, a, false, bf[g], (short)0, cinit, false, false);
                }
                v8f& cx = jt ? cx1 : cx0;
                v8f hxv;
                #pragma unroll
                for (int j = 0; j < 8; ++j) {
                    const float c2 = gate[0][j] * cx[j] + gate[1][j] * gate[2][j];
                    cx[j]  = c2;
                    hxv[j] = gate[3][j] * tanhf(c2);
                }
                v4f lo = { hxv[0], hxv[1], hxv[2], hxv[3] };
                v4f hh = { hxv[4], hxv[5], hxv[6], hxv[7] };
                // hx -> LDS for next step's GEMM1
                float* hp = &hx_s[n16 * HXS + h0 + 8 * hi];
                *(v4f*)hp       = lo;
                *(v4f*)(hp + 4) = hh;
                // hx -> outputs[t]
                float* op = out + (((size_t)t * B + gb0 + n16) * H + h0 + 8 * hi);
                *(v4f*)op       = lo;
                *(v4f*)(op + 4) = hh;
            }
        }
        __syncthreads();
    }

    // -------- final hx / cx --------
    const size_t hx_off = (size_t)S * B * H;
    for (int i = tid; i < NB * H; i += blockDim.x) {
        const int b = i >> 8, h = i & 255;
        out[hx_off + (size_t)(gb0 + b) * H + h] = hx_s[b * HXS + h];
    }
    const size_t cx_off = hx_off + (size_t)B * H;
    #pragma unroll
    for (int jt = 0; jt < 2; ++jt) {
        const int h0 = 32 * wv + 16 * jt;
        v8f& cx = jt ? cx1 : cx0;
        v4f lo = { cx[0], cx[1], cx[2], cx[3] };
        v4f hh = { cx[4], cx[5], cx[6], cx[7] };
        float* cp = out + cx_off + (size_t)(gb0 + n16) * H + h0 + 8 * hi;
        *(v4f*)cp       = lo;
        *(v4f*)(cp + 4) = hh;
    }
}

extern "C" void kernel_launch(void* const* d_in, const int* in_sizes, int n_in,
                              void* d_out, int out_size, void* d_ws, size_t ws_size,
                              hipStream_t stream) {
    (void)in_sizes; (void)n_in; (void)out_size; (void)d_ws; (void)ws_size;
    const float* inputs = (const float*)d_in[0];
    const float* w_lin  = (const float*)d_in[1];
    const float* b_lin  = (const float*)d_in[2];
    const float* w_map  = (const float*)d_in[3];
    const float* b_map  = (const float*)d_in[4];
    const float* theta  = (const float*)d_in[5];
    float* out = (float*)d_out;

    qlstm_scan_kernel<<<dim3(B / NB), dim3(256), 0, stream>>>(
        inputs, w_lin, b_lin, w_map, b_map, theta, out);
}